// ConLSTM_CRF_17884243820674
// MI455X (gfx1250) — compile-verified
//
#include <hip/hip_runtime.h>
#include <hip/hip_bf16.h>

// CRF forward + gold + Viterbi for B=128, T=4096, K=64 on gfx1250 (MI455X).
//
// Kernel 1 (forward / log-partition): exp-space GEMM formulation.
//   alpha' = log( E @ exp(alpha - m) ) + m + feat,  E = exp(transitions) fixed.
//   16 batches per block -> per-step dense GEMM M=64,N=16,K=64 via
//   V_WMMA_F32_16X16X4_F32 (4 waves x 16 chained WMMA, A-tiles resident in
//   VGPRs, B-tiles preloaded per step so the WMMA chain runs without
//   per-instruction dscnt stalls). Hardware v_exp_f32/v_log_f32 (base-2).
// Kernel 2 (gold score + nll): parallel gather-reduce.
// Kernel 3 (Viterbi): max-plus per-batch block; backpointers (256KB) live
//   entirely in the 320KB CDNA5 LDS; backtrack is an LDS pointer chase.

#define CRF_B 128
#define CRF_T 4096
#define CRF_K 64
#define CRF_START 62
#define CRF_STOP 63
#define CRF_NEG (-10000.0f)

#define L2E 1.4426950408889634f   // log2(e)
#define LN2 0.6931471805599453f   // ln(2)

typedef __attribute__((ext_vector_type(2))) float v2f;
typedef __attribute__((ext_vector_type(8))) float v8f;

#define AS 17   // padded row stride for alpha / X (bank-conflict avoidance)
#define FS 68   // padded row stride for staged feats

// ---------------------------------------------------------------------------
// Kernel 1: forward algorithm, 16 batches per block, 128 threads (4 waves).
// ---------------------------------------------------------------------------
__global__ __launch_bounds__(128) void crf_forward_kernel(
    const float* __restrict__ feats,       // [B][T][K]
    const float* __restrict__ trans,       // [K][K]  trans[next][prev]
    float* __restrict__ fwd_out)           // [B]
{
  __shared__ float alpha[CRF_K][AS];       // log-domain state, [i][n]
  __shared__ float Xs[CRF_K][AS];          // exp(alpha - m), [i][n]
  __shared__ float ms[16];                 // per-batch running max
  __shared__ float featL[2][16][FS];       // double-buffered feats [buf][n][j]

  const int tid  = threadIdx.x;
  const int lane = tid & 31;
  const int wave = tid >> 5;               // M-tile index (0..3)
  const int bg   = blockIdx.x;             // batch group (16 batches)

  // --- preload A fragments: E = exp(trans), tile (wave, kt), 16x4 f32 each.
  // A 16x4 layout: lanes 0-15 hold K={0,1}, lanes 16-31 hold K={2,3}.
  v2f afrag[16];
  {
    const int m  = wave * 16 + (lane & 15);
    const int kp = (lane >> 4) * 2;
#pragma unroll
    for (int kt = 0; kt < 16; ++kt) {
      afrag[kt].x = __builtin_amdgcn_exp2f(trans[m * CRF_K + kt * 4 + kp] * L2E);
      afrag[kt].y = __builtin_amdgcn_exp2f(trans[m * CRF_K + kt * 4 + kp + 1] * L2E);
    }
  }

  // --- init alpha: START = 0, else NEG
  for (int idx = tid; idx < CRF_K * 16; idx += 128) {
    const int i = idx >> 4, n = idx & 15;
    alpha[i][n] = (i == CRF_START) ? 0.0f : CRF_NEG;
  }

  // --- stage feats for t = 0 (each thread: one batch row, 8 floats)
  const int pn = tid >> 3, pq = tid & 7;
  {
    const float* p = feats + ((size_t)(bg * 16 + pn) * CRF_T) * CRF_K + pq * 8;
    *(float4*)&featL[0][pn][pq * 8]     = *(const float4*)(p);
    *(float4*)&featL[0][pn][pq * 8 + 4] = *(const float4*)(p + 4);
  }
  __syncthreads();

  const int n_red = tid >> 3, seg = tid & 7;   // m/X mapping: batch n_red
  const int n_c = lane & 15, hi_c = lane >> 4; // WMMA B/C fragment mapping

  for (int t = 0; t < CRF_T; ++t) {
    // issue global loads for t+1 early; land in LDS at end of step
    float4 f0, f1;
    if (t + 1 < CRF_T) {
      const float* p =
          feats + ((size_t)(bg * 16 + pn) * CRF_T + (t + 1)) * CRF_K + pq * 8;
      f0 = *(const float4*)(p);
      f1 = *(const float4*)(p + 4);
      __builtin_prefetch(p + 8 * CRF_K, 0, 1);   // ~8 steps ahead
    }

    // --- fused: per-batch max m[n] + X = exp2((alpha - m)*log2e)
    float av[8];
    float mx = -3.4e38f;
#pragma unroll
    for (int k2 = 0; k2 < 8; ++k2) {
      av[k2] = alpha[seg * 8 + k2][n_red];
      mx = fmaxf(mx, av[k2]);
    }
    mx = fmaxf(mx, __shfl_xor(mx, 1));
    mx = fmaxf(mx, __shfl_xor(mx, 2));
    mx = fmaxf(mx, __shfl_xor(mx, 4));   // all 8 lanes now hold m[n_red]
    if (seg == 0) ms[n_red] = mx;
#pragma unroll
    for (int k2 = 0; k2 < 8; ++k2)
      Xs[seg * 8 + k2][n_red] = __builtin_amdgcn_exp2f((av[k2] - mx) * L2E);
    __syncthreads();

    // --- preload all 16 B fragments (issue every ds_load before the chain)
    v2f bfrag[16];
#pragma unroll
    for (int kt = 0; kt < 16; ++kt) {
      const int r0 = kt * 4 + hi_c * 2;  // B 4x16: lanes 0-15 K={0,1}, 16-31 K={2,3}
      bfrag[kt].x = Xs[r0][n_c];
      bfrag[kt].y = Xs[r0 + 1][n_c];
    }

    // --- Y = E @ X via 16 chained f32 WMMAs (M-tile per wave)
    v8f c = {};
#pragma unroll
    for (int kt = 0; kt < 16; ++kt)
      c = __builtin_amdgcn_wmma_f32_16x16x4_f32(
          false, afrag[kt], false, bfrag[kt], (short)0, c, false, false);

    // --- alpha' = log(Y) + m + feat   (C layout: vgpr r, lanes0-15 M=r, 16-31 M=r+8)
    {
      const float mm = ms[n_c];
#pragma unroll
      for (int r = 0; r < 8; ++r) {
        const int j = wave * 16 + r + hi_c * 8;
        const float a =
            __builtin_amdgcn_logf(c[r]) * LN2 + mm + featL[t & 1][n_c][j];
        alpha[j][n_c] = fmaxf(a, -30000.0f);   // clamp -inf (unreachable states)
      }
    }

    // --- land prefetched feats for t+1
    if (t + 1 < CRF_T) {
      *(float4*)&featL[(t + 1) & 1][pn][pq * 8]     = f0;
      *(float4*)&featL[(t + 1) & 1][pn][pq * 8 + 4] = f1;
    }
    __syncthreads();
  }

  // --- forward_score[n] = logsumexp_j( alpha[j][n] + trans[STOP][j] )
  float vals[8];
  float mx = -3.4e38f;
#pragma unroll
  for (int k2 = 0; k2 < 8; ++k2) {
    const int j = seg * 8 + k2;
    vals[k2] = alpha[j][n_red] + trans[CRF_STOP * CRF_K + j];
    mx = fmaxf(mx, vals[k2]);
  }
  mx = fmaxf(mx, __shfl_xor(mx, 1));
  mx = fmaxf(mx, __shfl_xor(mx, 2));
  mx = fmaxf(mx, __shfl_xor(mx, 4));
  float s = 0.0f;
#pragma unroll
  for (int k2 = 0; k2 < 8; ++k2)
    s += __builtin_amdgcn_exp2f((vals[k2] - mx) * L2E);
  s += __shfl_xor(s, 1);
  s += __shfl_xor(s, 2);
  s += __shfl_xor(s, 4);
  if (seg == 0)
    fwd_out[bg * 16 + n_red] = __builtin_amdgcn_logf(s) * LN2 + mx;
}

// ---------------------------------------------------------------------------
// Kernel 2: gold-path score + nll = fwd - gold.  One block per batch.
// ---------------------------------------------------------------------------
__global__ __launch_bounds__(256) void crf_gold_kernel(
    const float* __restrict__ feats, const int* __restrict__ tags,
    const float* __restrict__ trans, const float* __restrict__ fwd,
    float* __restrict__ nll_out)
{
  __shared__ float red[256];
  const int b = blockIdx.x, tid = threadIdx.x;
  const int* tg = tags + (size_t)b * CRF_T;
  const float* fb = feats + (size_t)b * CRF_T * CRF_K;

  float s = 0.0f;
  for (int t = tid; t < CRF_T; t += 256) {
    const int cur  = tg[t];
    const int prev = (t == 0) ? CRF_START : tg[t - 1];
    s += trans[cur * CRF_K + prev] + fb[(size_t)t * CRF_K + cur];
  }
  if (tid == 0) s += trans[CRF_STOP * CRF_K + tg[CRF_T - 1]];

  red[tid] = s;
  __syncthreads();
  for (int st = 128; st > 0; st >>= 1) {
    if (tid < st) red[tid] += red[tid + st];
    __syncthreads();
  }
  if (tid == 0) nll_out[b] = fwd[b] - red[0];
}

// ---------------------------------------------------------------------------
// Kernel 3: Viterbi + in-LDS backtrack.  One block per batch, 256 threads.
// Dynamic LDS: bp[T][K] u8 (256KB) + trans[K][65] + delta ping-pong + feats.
// ---------------------------------------------------------------------------
__global__ __launch_bounds__(256) void crf_viterbi_kernel(
    const float* __restrict__ feats, const float* __restrict__ trans,
    float* __restrict__ path_out,      // [B][T]  (tags stored as float)
    float* __restrict__ score_out)     // [B]
{
  extern __shared__ char smem[];
  unsigned char* bp = (unsigned char*)smem;                         // [T][K]
  float* trans_s = (float*)(smem + (size_t)CRF_T * CRF_K);          // [K][65]
  float* delta0  = trans_s + CRF_K * 65;                            // [K]
  float* delta1  = delta0 + CRF_K;                                  // [K]
  float* featb   = delta1 + CRF_K;                                  // [2][K]

  const int tid = threadIdx.x;
  const int b   = blockIdx.x;
  const float* fb = feats + (size_t)b * CRF_T * CRF_K;

  // load transitions with padded stride (bank-conflict-free inner loop)
  for (int idx = tid; idx < CRF_K * CRF_K; idx += 256) {
    const int j = idx >> 6, i = idx & 63;
    trans_s[j * 65 + i] = trans[idx];
  }
  if (tid < CRF_K) delta0[tid] = (tid == CRF_START) ? 0.0f : CRF_NEG;
  if (tid < 16) *(float4*)&featb[tid * 4] = *(const float4*)(fb + tid * 4);
  __syncthreads();

  const int j = tid >> 2, q = tid & 3;   // 4 lanes per next-tag j
  float* dcur = delta0;
  float* dnxt = delta1;

  for (int t = 0; t < CRF_T; ++t) {
    float4 fpre;
    if (tid < 16 && t + 1 < CRF_T) {
      fpre = *(const float4*)(fb + (size_t)(t + 1) * CRF_K + tid * 4);
      __builtin_prefetch(fb + (size_t)(t + 8) * CRF_K + tid * 4, 0, 1);
    }

    // local argmax over 16 prev-tags
    float best = -3.4e38f;
    int bi = 0;
    const float* trow = trans_s + j * 65 + q * 16;
#pragma unroll
    for (int k = 0; k < 16; ++k) {
      const float sc = dcur[q * 16 + k] + trow[k];
      if (sc > best) { best = sc; bi = q * 16 + k; }
    }
    // combine the 4 lanes of this j (wave32 shuffles); ties -> lowest index
#pragma unroll
    for (int off = 1; off < 4; off <<= 1) {
      const float ob = __shfl_xor(best, off);
      const int   oi = __shfl_xor(bi, off);
      if (ob > best || (ob == best && oi < bi)) { best = ob; bi = oi; }
    }
    if (q == 0) {
      bp[t * CRF_K + j] = (unsigned char)bi;
      dnxt[j] = best + featb[(t & 1) * CRF_K + j];
    }
    if (tid < 16 && t + 1 < CRF_T)
      *(float4*)&featb[((t + 1) & 1) * CRF_K + tid * 4] = fpre;
    __syncthreads();
    float* tmp = dcur; dcur = dnxt; dnxt = tmp;
  }

  // terminal scores + argmax
  if (tid < CRF_K) dnxt[tid] = dcur[tid] + trans_s[CRF_STOP * 65 + tid];
  __syncthreads();
  if (tid < 32) {
    float v = dnxt[tid];
    int vi = tid;
    const float v2 = dnxt[tid + 32];
    if (v2 > v) { v = v2; vi = tid + 32; }
#pragma unroll
    for (int off = 1; off < 32; off <<= 1) {
      const float ov = __shfl_xor(v, off);
      const int   oi = __shfl_xor(vi, off);
      if (ov > v || (ov == v && oi < vi)) { v = ov; vi = oi; }
    }
    if (tid == 0) {
      score_out[b] = v;
      int tag = vi;
      float* po = path_out + (size_t)b * CRF_T;
      for (int t = CRF_T - 1; t >= 0; --t) {
        po[t] = (float)tag;
        tag = bp[t * CRF_K + tag];
      }
    }
  }
}

// ---------------------------------------------------------------------------
extern "C" void kernel_launch(void* const* d_in, const int* in_sizes, int n_in,
                              void* d_out, int out_size, void* d_ws,
                              size_t ws_size, hipStream_t stream) {
  const float* feats = (const float*)d_in[0];   // [B][T][K] f32
  const int*   tags  = (const int*)d_in[1];     // [B][T]    i32
  const float* trans = (const float*)d_in[2];   // [K][K]    f32

  float* out   = (float*)d_out;
  float* nll   = out;                  // [B]
  float* score = out + CRF_B;          // [B]
  float* path  = out + 2 * CRF_B;      // [B][T]
  float* fwdws = (float*)d_ws;         // [B] forward scores (scratch)

  crf_forward_kernel<<<CRF_B / 16, 128, 0, stream>>>(feats, trans, fwdws);
  crf_gold_kernel<<<CRF_B, 256, 0, stream>>>(feats, tags, trans, fwdws, nll);

  const size_t dynLDS = (size_t)CRF_T * CRF_K            // bp u8
                      + (size_t)CRF_K * 65 * sizeof(float)
                      + 2 * CRF_K * sizeof(float)
                      + 2 * CRF_K * sizeof(float);       // ~273.3 KB < 320 KB
  crf_viterbi_kernel<<<CRF_B, 256, dynLDS, stream>>>(feats, trans, path, score);
}